// DistLoss_52561809768921
// MI455X (gfx1250) — compile-verified
//
#include <hip/hip_runtime.h>
#include <hip/hip_bf16.h>
#include <math.h>

typedef float v2f __attribute__((ext_vector_type(2)));
typedef float v8f __attribute__((ext_vector_type(8)));

#define Bsz 4
#define Csz 16
#define Hsz 256
#define Wsz 256
#define HW  (Hsz * Wsz)
#define NUMEL (6553600LL)   // B*H*W*25

// staged x tile geometry: 18 rows x 26 valid cols, padded row stride 27
#define SXROWS 18
#define SXCOLS 26
#define SXW    27
#define SXCH   (SXROWS * SXW)   // 486 floats per channel

// 25 neighbor offsets, sorted by flattened 11x11 window position (dy, then dx)
__constant__ int c_dy[25] = {-5,-5,-5,-2,-2,-2,-1,-1,-1, 0, 0, 0, 0, 0, 0, 0, 1, 1, 1, 2, 2, 2, 5, 5, 5};
__constant__ int c_dx[25] = {-5, 0, 5,-2, 0, 2,-1, 0, 1,-5,-2,-1, 0, 1, 2, 5,-1, 0, 1,-2, 0, 2,-5, 0, 5};

__device__ __forceinline__ int refl(int i) {            // np.pad mode='reflect', H==W==256
    i = (i < 0) ? -i : i;
    return (i >= Hsz) ? (2 * Hsz - 2 - i) : i;
}

__device__ __forceinline__ float gelu_exact(float v) {
    return 0.5f * v * (1.0f + erff(v * 0.7071067811865475f));
}

__global__ void dl_init_ws(double* wsum, unsigned int* wcnt) {
    wsum[0] = 0.0; wsum[1] = 0.0; *wcnt = 0u;
}

__global__ __launch_bounds__(256) void dl_main(
    const float* __restrict__ x, const int* __restrict__ y,
    const float* __restrict__ W1, const float* __restrict__ b1,
    const float* __restrict__ W2, const float* __restrict__ b2,
    float* __restrict__ out, double* __restrict__ wsum, unsigned int* __restrict__ wcnt)
{
    __shared__ float s_x[Csz * SXCH];   // 31.1 KB: x halo tile (reflection resolved)
    __shared__ int   s_y[SXCH];         // 1.9 KB : label halo tile
    __shared__ float s_h[8 * 256];      // 8 KB   : per-wave 16x16 transpose bounce
    __shared__ float s_w1[28 * 16];     // zero-padded W1  (K=28, N=16)
    __shared__ float s_w2[12 * 32];     // zero-padded W2  (K=12, N=32)
    __shared__ float s_b1[16];
    __shared__ float s_b2[32];

    const int tid  = threadIdx.x;
    const int wave = tid >> 5;          // row within 8x16 pixel tile
    const int lane = tid & 31;
    const bool lo  = (lane < 16);
    const int m    = lane & 15;         // pixel col in tile / N column for WMMA

    // block -> (batch, 8-row tile, 16-col tile)
    const int blk   = blockIdx.x;
    const int b     = blk >> 9;                 // / 512
    const int rem   = blk & 511;
    const int hB    = (rem >> 4) << 3;          // tile row base
    const int wB    = (rem & 15) << 4;          // tile col base

    // ---------------- cooperative staging into LDS ----------------
    const float* xb = x + (size_t)b * Csz * HW;
    for (int idx = tid; idx < Csz * SXROWS * SXCOLS; idx += 256) {
        const int c  = idx / (SXROWS * SXCOLS);
        const int rc = idx - c * (SXROWS * SXCOLS);
        const int r  = rc / SXCOLS;
        const int cc = rc - r * SXCOLS;
        const int gh = refl(hB + r - 5);
        const int gw = refl(wB + cc - 5);
        s_x[c * SXCH + r * SXW + cc] = xb[(size_t)c * HW + gh * Wsz + gw];
    }
    const int* yg = y + b * HW;
    for (int idx = tid; idx < SXROWS * SXCOLS; idx += 256) {
        const int r  = idx / SXCOLS;
        const int cc = idx - r * SXCOLS;
        s_y[r * SXW + cc] = yg[refl(hB + r - 5) * Wsz + refl(wB + cc - 5)];
    }
    for (int idx = tid; idx < 28 * 16; idx += 256) {
        const int k = idx >> 4, n = idx & 15;
        s_w1[idx] = (k < 25 && n < 10) ? W1[k * 10 + n] : 0.0f;
    }
    for (int idx = tid; idx < 12 * 32; idx += 256) {
        const int k = idx >> 5, n = idx & 31;
        s_w2[idx] = (k < 10 && n < 25) ? W2[k * 25 + n] : 0.0f;
    }
    if (tid < 16)                 s_b1[tid] = (tid < 10) ? b1[tid] : 0.0f;
    else if (tid < 48) { const int n = tid - 16; s_b2[n] = (n < 25) ? b2[n] : 0.0f; }
    __syncthreads();

    // ---------------- L1 distances from LDS tile ----------------
    const int j = wave;
    const float* sxc = s_x + (5 + j) * SXW + (5 + m);   // center, channel 0
    float xc[Csz];
    #pragma unroll
    for (int c = 0; c < Csz; ++c) xc[c] = sxc[c * SXCH];

    float d[13];
    #pragma unroll
    for (int it = 0; it < 13; ++it) {
        const int n = 2 * it + (lo ? 0 : 1);            // lanes<16 even n, lanes>=16 odd n
        float dv = 0.0f;
        if (n < 25) {
            const float* sxn = s_x + (5 + j + c_dy[n]) * SXW + (5 + m + c_dx[n]);
            #pragma unroll
            for (int c = 0; c < Csz; ++c) dv += fabsf(xc[c] - sxn[c * SXCH]);
        }
        d[it] = dv;
    }

    // ---------------- layer 1: [16x28] @ [28x16] via 7x WMMA f32 16x16x4 ----------------
    v8f acc = {0.f,0.f,0.f,0.f,0.f,0.f,0.f,0.f};
    #pragma unroll
    for (int kk = 0; kk < 7; ++kk) {
        const int k0 = 4 * kk + (lo ? 0 : 2);           // B layout: half-wave holds rows K0/K0+1
        v2f bm;
        bm.x = s_w1[k0 * 16 + m];
        bm.y = s_w1[(k0 + 1) * 16 + m];
        v2f am;
        if (kk < 6) {
            const float e  = d[2 * kk];                 // n = 4kk   (even half)
            const float o  = d[2 * kk + 1];             // n = 4kk+3 (odd half)
            const float t0 = __shfl_xor(e, 16, 32);
            const float t1 = __shfl_xor(o, 16, 32);
            am.x = lo ? e  : t1;                        // K = 4kk   | 4kk+2
            am.y = lo ? t0 : o;                         // K = 4kk+1 | 4kk+3
        } else {                                        // K = 24..27: only n=24 real
            am.x = lo ? d[12] : 0.0f;
            am.y = 0.0f;
        }
        acc = __builtin_amdgcn_wmma_f32_16x16x4_f32(false, am, false, bm,
                                                    (short)0, acc, false, false);
    }

    // ---------------- bias + exact GELU, transpose D-layout -> A-layout via LDS ----------------
    float* sh = s_h + wave * 256;
    const float bb1 = s_b1[m];
    #pragma unroll
    for (int i = 0; i < 8; ++i) {
        const int M = i + (lo ? 0 : 8);
        sh[M * 16 + m] = gelu_exact(acc[i] + bb1);      // cols >=10 stay exactly 0
    }
    __syncthreads();

    // ---------------- layer 2: [16x12] @ [12x32] via 2 x 3 WMMAs ----------------
    v8f acc0 = {0.f,0.f,0.f,0.f,0.f,0.f,0.f,0.f};
    v8f acc1 = {0.f,0.f,0.f,0.f,0.f,0.f,0.f,0.f};
    #pragma unroll
    for (int kk = 0; kk < 3; ++kk) {
        const int k0 = 4 * kk + (lo ? 0 : 2);
        v2f am;
        am.x = sh[m * 16 + k0];
        am.y = sh[m * 16 + k0 + 1];
        v2f bm0, bm1;
        bm0.x = s_w2[k0 * 32 + m];
        bm0.y = s_w2[(k0 + 1) * 32 + m];
        bm1.x = s_w2[k0 * 32 + 16 + m];
        bm1.y = s_w2[(k0 + 1) * 32 + 16 + m];
        acc0 = __builtin_amdgcn_wmma_f32_16x16x4_f32(false, am, false, bm0,
                                                     (short)0, acc0, false, false);
        acc1 = __builtin_amdgcn_wmma_f32_16x16x4_f32(false, am, false, bm1,
                                                     (short)0, acc1, false, false);
    }

    // ---------------- bias + ReLU, store dist_l1, accumulate loss terms ----------------
    const int   n0 = m, n1 = 16 + m;
    const float bias20 = s_b2[n0];
    const float bias21 = s_b2[n1 & 31];
    const int dy0 = c_dy[n0],                  dx0 = c_dx[n0];
    const int dy1 = (n1 < 25) ? c_dy[n1] : 0,  dx1 = (n1 < 25) ? c_dx[n1] : 0;
    const int rowPix = (b * Hsz + hB + j) * Wsz + wB;   // flat pixel id of (row, wB)

    float sm = 0.0f, sx = 0.0f;
    int cnt = 0;
    #pragma unroll
    for (int i = 0; i < 8; ++i) {
        const int M  = i + (lo ? 0 : 8);
        const int yc = s_y[(5 + j) * SXW + 5 + M];

        const float r0 = fmaxf(acc0[i] + bias20, 0.0f);
        out[(size_t)(rowPix + M) * 25 + n0] = r0;
        if (s_y[(5 + j + dy0) * SXW + 5 + M + dx0] == yc) { ++cnt; sm += fmaxf(r0 - 0.5f, 0.0f); }
        else                                              {        sx += fmaxf(2.0f - r0, 0.0f); }

        if (n1 < 25) {
            const float r1 = fmaxf(acc1[i] + bias21, 0.0f);
            out[(size_t)(rowPix + M) * 25 + n1] = r1;
            if (s_y[(5 + j + dy1) * SXW + 5 + M + dx1] == yc) { ++cnt; sm += fmaxf(r1 - 0.5f, 0.0f); }
            else                                              {        sx += fmaxf(2.0f - r1, 0.0f); }
        }
    }

    // ---------------- wave reduction + global atomics ----------------
    #pragma unroll
    for (int off = 16; off >= 1; off >>= 1) {
        sm  += __shfl_xor(sm, off, 32);
        sx  += __shfl_xor(sx, off, 32);
        cnt += __shfl_xor(cnt, off, 32);
    }
    if (lane == 0) {
        atomicAdd(&wsum[0], (double)sm);
        atomicAdd(&wsum[1], (double)sx);
        atomicAdd(wcnt, (unsigned int)cnt);
    }
}

__global__ void dl_finalize(const double* __restrict__ wsum,
                            const unsigned int* __restrict__ wcnt,
                            float* __restrict__ out)
{
    const double npos = (double)(*wcnt);
    const double nneg = (double)NUMEL - npos;
    const double loss = (wsum[0] * nneg + wsum[1] * npos) /
                        ((double)NUMEL * (double)NUMEL);
    out[NUMEL] = (float)loss;
}

extern "C" void kernel_launch(void* const* d_in, const int* in_sizes, int n_in,
                              void* d_out, int out_size, void* d_ws, size_t ws_size,
                              hipStream_t stream) {
    const float* x  = (const float*)d_in[0];
    const int*   y  = (const int*)  d_in[1];
    const float* W1 = (const float*)d_in[2];
    const float* b1 = (const float*)d_in[3];
    const float* W2 = (const float*)d_in[4];
    const float* b2 = (const float*)d_in[5];
    float* out = (float*)d_out;

    double*       wsum = (double*)d_ws;
    unsigned int* wcnt = (unsigned int*)((char*)d_ws + 16);

    dl_init_ws<<<1, 1, 0, stream>>>(wsum, wcnt);
    // B * (H/8) * (W/16) = 4 * 32 * 16 = 2048 blocks of 256 threads (8 waves)
    dl_main<<<2048, 256, 0, stream>>>(x, y, W1, b1, W2, b2, out, wsum, wcnt);
    dl_finalize<<<1, 1, 0, stream>>>(wsum, wcnt, out);
}